// SlidingWindowMatrixMult3D_73452530696595
// MI455X (gfx1250) — compile-verified
//
#include <hip/hip_runtime.h>
#include <hip/hip_bf16.h>

// Z[b,w,t] = sum_c Q[b,w,c]*K[b,w+t,c] + sum_c Q[b,w,c]*bias[c,t]
// B=16, T=1024, C=128. All f32; computed with V_WMMA_F32_16X16X4_F32.

typedef float v2f __attribute__((ext_vector_type(2)));
typedef float v8f __attribute__((ext_vector_type(8)));

#define BB 16
#define TT 1024
#define CC 128
#define KTT (2*TT - 1)   // 2047 key positions

// One wave handles one (b, w-tile) x 4 consecutive t-tiles.
// Block = 4 waves (128 threads) covering 4 wave-slots in t.
// grid.x = BB * (TT/16) * ((TT/16)/(4 waves * 4 tiles)) = 16 * 64 * 4 = 4096

__global__ __launch_bounds__(128)
void swin_qkbias_kernel(const float* __restrict__ Q,
                        const float* __restrict__ K,
                        const float* __restrict__ bias,
                        float* __restrict__ out) {
    // per-wave LDS staging for diagonal gather: 16 rows x 32 cols, stride 36
    __shared__ float lds[4][16 * 36];

    const int lane = threadIdx.x & 31;
    const int wv   = threadIdx.x >> 5;   // wave in block, 0..3
    const int half = lane >> 4;          // 0: lanes 0-15, 1: lanes 16-31
    const int l16  = lane & 15;

    const int bid   = blockIdx.x;
    const int tgrp  = bid & 3;           // 4 t-groups of 256 columns
    const int wtile = (bid >> 2) & 63;   // 64 w-tiles
    const int b     = bid >> 8;          // batch 0..15

    const int w0      = wtile * 16;
    const int t0_base = tgrp * 256 + wv * 64;   // this wave: t-tiles t0_base + 16q, q=0..3

    const float* Qb = Q + (size_t)b * TT * CC;
    const float* Kb = K + (size_t)b * KTT * CC;
    float*       Ob = out + (size_t)b * TT * TT;
    float* plds = &lds[wv][0];

    // ---- Preload Q tile (16x128) as 32 A-fragments (A: 16x4, M=l16, K=v+2*half) ----
    v2f qa[32];
    {
        const float* qrow = Qb + (size_t)(w0 + l16) * CC + 2 * half;
        #pragma unroll
        for (int n = 0; n < 32; ++n)
            qa[n] = *(const v2f*)(qrow + 4 * n);
    }

    const int base0 = w0 + t0_base;   // key-row base for this wave's band

    // ---- P_0 = Q_tile @ K[base0 .. base0+15]^T ----
    v8f p_cur;
    {
        int row = base0 + l16;
        if (row > KTT - 1) row = KTT - 1;   // clamp (clamped row's products never used)
        const float* krow = Kb + (size_t)row * CC + 2 * half;
        v8f p = {};
        #pragma unroll
        for (int n = 0; n < 32; ++n) {
            v2f bv = *(const v2f*)(krow + 4 * n);
            p = __builtin_amdgcn_wmma_f32_16x16x4_f32(false, qa[n], false, bv,
                                                      (short)0, p, false, false);
        }
        p_cur = p;
    }

    for (int q = 0; q < 4; ++q) {
        const int t0 = t0_base + 16 * q;

        // ---- P_{q+1} = Q_tile @ K[base0+16(q+1) .. +15]^T ----
        v8f p_next;
        {
            int row = base0 + 16 * (q + 1) + l16;
            if (row > KTT - 1) row = KTT - 1;
            const float* krow = Kb + (size_t)row * CC + 2 * half;
            v8f p = {};
            #pragma unroll
            for (int n = 0; n < 32; ++n) {
                v2f bv = *(const v2f*)(krow + 4 * n);
                p = __builtin_amdgcn_wmma_f32_16x16x4_f32(false, qa[n], false, bv,
                                                          (short)0, p, false, false);
            }
            p_next = p;
        }

        // ---- bias GEMM: acc = Q_tile @ bias[:, t0..t0+15] ----
        v8f acc = {};
        {
            const float* bcol = bias + (size_t)(2 * half) * TT + t0 + l16;
            #pragma unroll
            for (int n = 0; n < 32; ++n) {
                v2f bv;
                bv.x = bcol[(size_t)(4 * n) * TT];
                bv.y = bcol[(size_t)(4 * n + 1) * TT];
                acc = __builtin_amdgcn_wmma_f32_16x16x4_f32(false, qa[n], false, bv,
                                                            (short)0, acc, false, false);
            }
        }

        // ---- stage P_q | P_{q+1} in LDS (16x32, stride 36), gather Z_k[i,j] = P[i, i+j] ----
        asm volatile("" ::: "memory");
        #pragma unroll
        for (int r = 0; r < 8; ++r) {
            int row = r + 8 * half;                 // C/D layout: M = r + 8*(lane/16)
            plds[row * 36 + l16]      = p_cur[r];   // cols 0..15  = P_q
            plds[row * 36 + 16 + l16] = p_next[r];  // cols 16..31 = P_{q+1}
        }
        asm volatile("s_wait_dscnt 0x0" ::: "memory");
        #pragma unroll
        for (int r = 0; r < 8; ++r) {
            int row = r + 8 * half;
            acc[r] += plds[row * 36 + row + l16];   // col = i + j, i = row, j = l16
        }
        asm volatile("" ::: "memory");

        // ---- store output tile (each VGPR half: contiguous 16-float row segment) ----
        #pragma unroll
        for (int r = 0; r < 8; ++r) {
            int row = r + 8 * half;
            Ob[(size_t)(w0 + row) * TT + t0 + l16] = acc[r];
        }

        p_cur = p_next;
    }
}

extern "C" void kernel_launch(void* const* d_in, const int* in_sizes, int n_in,
                              void* d_out, int out_size, void* d_ws, size_t ws_size,
                              hipStream_t stream) {
    (void)in_sizes; (void)n_in; (void)d_ws; (void)ws_size; (void)out_size;
    const float* Q    = (const float*)d_in[0];
    const float* K    = (const float*)d_in[1];
    const float* bias = (const float*)d_in[2];
    float* out = (float*)d_out;

    dim3 grid(BB * 64 * 4);   // 4096 blocks
    dim3 block(128);          // 4 waves of 32
    swin_qkbias_kernel<<<grid, block, 0, stream>>>(Q, K, bias, out);
}